// SharedTverskyCompact_5944234738210
// MI455X (gfx1250) — compile-verified
//
#include <hip/hip_runtime.h>
#include <cstdint>

#define BDIM 512
#define PDIM 512
#define FDIM 1024
#define TB 64            // output tile rows (B)
#define TP 32            // output tile cols (P)
#define FC 64            // F chunk resident in LDS
#define NC (FDIM / FC)   // 16 chunks total
#define SPLIT 2          // F-dimension split across blockIdx.z
#define NCS (NC / SPLIT) // 8 chunks per block

// ---- CDNA5 async global->LDS copy (ASYNCcnt-tracked) -----------------------
__device__ __forceinline__ void async_copy_b128(uint32_t lds_byte, const float* gsrc) {
  asm volatile("global_load_async_to_lds_b128 %0, %1, off"
               :: "v"(lds_byte), "v"(gsrc)
               : "memory");
}
__device__ __forceinline__ void wait_async0() {
  asm volatile("s_wait_asynccnt 0x0" ::: "memory");
}

// ---- Pass 1: sigmoid + transpose  (dst[n][m] = sigmoid(src[m][n])) ---------
__global__ __launch_bounds__(256) void sig_transpose_k(const float* __restrict__ src,
                                                       float* __restrict__ dst,
                                                       int M, int N) {
  __shared__ float tile[32][33];
  const int bx = blockIdx.x * 32;  // N index
  const int by = blockIdx.y * 32;  // M index
  const int tx = threadIdx.x;      // 0..31
  const int ty = threadIdx.y;      // 0..7
#pragma unroll
  for (int k = 0; k < 32; k += 8) {
    float v = src[(size_t)(by + ty + k) * N + bx + tx];
    tile[ty + k][tx] = 1.0f / (1.0f + expf(-v));
  }
  __syncthreads();
#pragma unroll
  for (int k = 0; k < 32; k += 8) {
    dst[(size_t)(bx + ty + k) * M + by + tx] = tile[tx][ty + k];
  }
}

// ---- Pass 1b: per-row sigmoid sums (column sums of transposed layout) ------
// 4 independent accumulators break the serial-add dependency chain.
__global__ __launch_bounds__(256) void colsum_k(const float* __restrict__ xT,
                                                const float* __restrict__ wT,
                                                float* __restrict__ Sx,
                                                float* __restrict__ Sw) {
  const int g = blockIdx.x * 256 + threadIdx.x;   // 0..1023
  const float* src = (g < BDIM) ? xT : wT;
  const int col = g & (BDIM - 1);
  float s0 = 0.f, s1 = 0.f, s2 = 0.f, s3 = 0.f;
  for (int f = 0; f < FDIM; f += 4) {
    s0 += src[(size_t)(f + 0) * BDIM + col];
    s1 += src[(size_t)(f + 1) * BDIM + col];
    s2 += src[(size_t)(f + 2) * BDIM + col];
    s3 += src[(size_t)(f + 3) * BDIM + col];
  }
  const float s = (s0 + s1) + (s2 + s3);
  if (g < BDIM) Sx[col] = s; else Sw[col] = s;
}

// ---- Pass 2: partial pairwise L1 distance over one F-split -----------------
__global__ __launch_bounds__(256) void tversky_A_k(const float* __restrict__ xT,
                                                   const float* __restrict__ wT,
                                                   float* __restrict__ Apart) {
  __shared__ float xbuf[2][FC][TB];   // 2 * 16 KB
  __shared__ float wbuf[2][FC][TP];   // 2 *  8 KB
  const int tid   = threadIdx.x;
  const int tx    = tid & 15;         // P direction (16 * 2 = 32)
  const int ty    = tid >> 4;         // B direction (16 * 4 = 64)
  const int tileP = blockIdx.x * TP;
  const int tileB = blockIdx.y * TB;
  const int z     = blockIdx.z;       // F-split id
  const int c0    = z * NCS;

  auto issue = [&](int c, int bb) {
    const int cbase = c * FC;
    const uint32_t xb = (uint32_t)(uintptr_t)(&xbuf[bb][0][0]);
    const uint32_t wb = (uint32_t)(uintptr_t)(&wbuf[bb][0][0]);
    // x tile: FC x TB floats = 1024 x 16B transfers, 4 per thread
#pragma unroll
    for (int k = 0; k < 4; ++k) {
      const int i  = tid + k * 256;
      const int f  = i >> 4;
      const int cg = (i & 15) << 2;
      async_copy_b128(xb + (uint32_t)((f * TB + cg) * 4),
                      xT + (size_t)(cbase + f) * BDIM + tileB + cg);
    }
    // w tile: FC x TP floats = 512 x 16B transfers, 2 per thread
#pragma unroll
    for (int k = 0; k < 2; ++k) {
      const int i  = tid + k * 256;
      const int f  = i >> 3;
      const int cg = (i & 7) << 2;
      async_copy_b128(wb + (uint32_t)((f * TP + cg) * 4),
                      wT + (size_t)(cbase + f) * PDIM + tileP + cg);
    }
  };

  float acc[4][2];
#pragma unroll
  for (int i = 0; i < 4; ++i) { acc[i][0] = 0.0f; acc[i][1] = 0.0f; }

  issue(c0, 0);
  wait_async0();
  __syncthreads();

  for (int c = 0; c < NCS; ++c) {
    const int bb = c & 1;
    if (c + 1 < NCS) issue(c0 + c + 1, bb ^ 1);  // overlap next copy with compute
    const float* xs = &xbuf[bb][0][0];
    const float* ws = &wbuf[bb][0][0];
#pragma unroll 8
    for (int f = 0; f < FC; ++f) {
      const float4 xv = *(const float4*)(xs + f * TB + ty * 4);  // ds_load_b128, broadcast
      const float2 wv = *(const float2*)(ws + f * TP + tx * 2);  // ds_load_b64, conflict-free
      const float xr[4] = {xv.x, xv.y, xv.z, xv.w};
      const float wr[2] = {wv.x, wv.y};
#pragma unroll
      for (int i2 = 0; i2 < 4; ++i2)
#pragma unroll
        for (int j = 0; j < 2; ++j)
          acc[i2][j] += fabsf(xr[i2] - wr[j]);   // v_sub + v_add |.| modifier
    }
    wait_async0();
    __syncthreads();
  }

  // Write partial A tile (disjoint per split -> deterministic)
  float* Az = Apart + (size_t)z * BDIM * PDIM;
  const int b0 = tileB + ty * 4;
  const int p0 = tileP + tx * 2;
#pragma unroll
  for (int i2 = 0; i2 < 4; ++i2) {
    float2 o; o.x = acc[i2][0]; o.y = acc[i2][1];
    *(float2*)(Az + (size_t)(b0 + i2) * PDIM + p0) = o;
  }
}

// ---- Pass 3: combine splits + Tversky epilogue -----------------------------
__global__ __launch_bounds__(256) void tversky_epi_k(const float* __restrict__ Apart,
                                                     const float* __restrict__ Sx,
                                                     const float* __restrict__ Sw,
                                                     const float* __restrict__ bias,
                                                     const float* __restrict__ alpha_p,
                                                     const float* __restrict__ beta_p,
                                                     float* __restrict__ out) {
  const int idx = blockIdx.x * 256 + threadIdx.x;  // 0..65535, 4 outputs each
  const int b = idx >> 7;                          // 128 quads per row
  const int p = (idx & 127) << 2;
  const float4 a0 = *(const float4*)(Apart + (size_t)b * PDIM + p);
  const float4 a1 = *(const float4*)(Apart + (size_t)BDIM * PDIM + (size_t)b * PDIM + p);
  const float4 sw = *(const float4*)(Sw + p);
  const float4 bi = *(const float4*)(bias + p);
  const float sx    = Sx[b];
  const float alpha = alpha_p[0];
  const float beta  = beta_p[0];

  const float Aq[4]  = {a0.x + a1.x, a0.y + a1.y, a0.z + a1.z, a0.w + a1.w};
  const float swq[4] = {sw.x, sw.y, sw.z, sw.w};
  const float biq[4] = {bi.x, bi.y, bi.z, bi.w};
  float oq[4];
#pragma unroll
  for (int j = 0; j < 4; ++j) {
    const float A     = Aq[j];
    const float inter = 0.5f * (sx + swq[j] - A);
    const float xd    = 0.5f * (A + sx - swq[j]);
    const float wd    = 0.5f * (A - sx + swq[j]);
    oq[j] = inter / (inter + alpha * xd + beta * wd + 1e-8f) + biq[j];
  }
  float4 o; o.x = oq[0]; o.y = oq[1]; o.z = oq[2]; o.w = oq[3];
  *(float4*)(out + (size_t)b * PDIM + p) = o;
}

extern "C" void kernel_launch(void* const* d_in, const int* in_sizes, int n_in,
                              void* d_out, int out_size, void* d_ws, size_t ws_size,
                              hipStream_t stream) {
  const float* x     = (const float*)d_in[0];  // (512, 1024)
  const float* w     = (const float*)d_in[1];  // (512, 1024)
  const float* bias  = (const float*)d_in[2];  // (512,)
  const float* alpha = (const float*)d_in[3];  // scalar
  const float* beta  = (const float*)d_in[4];  // scalar
  float* ws    = (float*)d_ws;
  float* xT    = ws;                 // 1024*512 floats: sigmoid(x)^T  [F][B]
  float* wT    = ws + 524288;        // 1024*512 floats: sigmoid(w)^T  [F][P]
  float* Sx    = ws + 1048576;       // 512 floats
  float* Sw    = ws + 1049088;       // 512 floats
  float* Apart = ws + 1049600;       // SPLIT * 512*512 floats: partial L1 sums
  float* out   = (float*)d_out;      // (512, 512)

  const dim3 tb(32, 8);
  sig_transpose_k<<<dim3(FDIM / 32, BDIM / 32), tb, 0, stream>>>(x, xT, BDIM, FDIM);
  sig_transpose_k<<<dim3(FDIM / 32, PDIM / 32), tb, 0, stream>>>(w, wT, PDIM, FDIM);
  colsum_k<<<4, 256, 0, stream>>>(xT, wT, Sx, Sw);
  tversky_A_k<<<dim3(PDIM / TP, BDIM / TB, SPLIT), 256, 0, stream>>>(xT, wT, Apart);
  tversky_epi_k<<<(BDIM * PDIM / 4) / 256, 256, 0, stream>>>(Apart, Sx, Sw, bias,
                                                             alpha, beta, out);
}